// LocSE_34608846471727
// MI455X (gfx1250) — compile-verified
//
#include <hip/hip_runtime.h>
#include <math.h>

// Problem constants (match reference)
#define Bb    4
#define Nn    16384
#define K1    17
#define CHs   53          // rppe channels
#define CHP   56          // padded to multiple of 4 (14 k-steps)
#define Uh    64          // UNITS/2
#define OUTC  128
#define PTS   16          // points per block -> 16*17 = 272 rows = 17 M-tiles
#define ROWS  (PTS * K1)  // 272
#define MTILE (ROWS / 16) // 17
#define NTILE (Uh / 16)   // 4
#define KSTEP (CHP / 4)   // 14
#define EPSF  1e-8f

typedef __attribute__((ext_vector_type(2))) float v2f;
typedef __attribute__((ext_vector_type(4))) float v4f;
typedef __attribute__((ext_vector_type(8))) float v8f;

__global__ __launch_bounds__(256) void locse_wmma_kernel(
    const float* __restrict__ pc,     // (B,N,2)
    const float* __restrict__ feats,  // (B,N,64)
    const int*   __restrict__ nidx,   // (B,N,17)
    const float* __restrict__ W,      // (53,64)
    const float* __restrict__ bias,   // (64,)
    float*       __restrict__ out,    // (B,N,17,128)
    float*       __restrict__ ggf)    // (B,N,1,2)
{
    extern __shared__ float smem[];
    float* ldsA  = smem;                   // ROWS*CHP rppe (A matrix, row major, K-padded)
    float* ldsWt = ldsA + ROWS * CHP;      // Uh*CHP  W transposed: Wt[n][k] (K-padded)
    float* cx    = ldsWt + Uh * CHP;       // ROWS neighbor x
    float* cy    = cx + ROWS;              // ROWS neighbor y
    float* nr    = cy + ROWS;              // ROWS neighbor norms
    int*   sidx  = (int*)(nr + ROWS);      // ROWS gathered indices

    const int tid        = threadIdx.x;
    const int blockStart = blockIdx.x * PTS;        // first global point of block
    const int b          = blockStart / Nn;         // batch (blocks never straddle)
    const int nbase      = blockStart % Nn;

    // ---- phase 0: gather neighbor coords / indices / norms ----
    for (int t = tid; t < ROWS; t += 256) {
        int lp = t / K1, j = t - lp * K1;
        int id = nidx[(size_t)(b * Nn + nbase + lp) * K1 + j];
        const float* p = pc + (size_t)(b * Nn + id) * 2;
        float px = p[0], py = p[1];
        cx[t] = px; cy[t] = py;
        nr[t] = sqrtf(px * px + py * py);
        sidx[t] = id;
    }
    // transposed, K-padded W into LDS: B-fragment pairs become one ds_load_b64
    for (int i = tid; i < Uh * CHP; i += 256) {
        int ncol = i / CHP, k = i - ncol * CHP;
        ldsWt[i] = (k < CHs) ? W[k * Uh + ncol] : 0.0f;
    }
    __syncthreads();

    // ---- phase 1a: build rppe rows in LDS ----
    for (int r = tid; r < ROWS; r += 256) {
        int lp = r / K1;
        float pxj = cx[r], pyj = cy[r], nj = nr[r];
        float* Ar = ldsA + r * CHP;
        Ar[0] = pxj; Ar[1] = pyj;
        int base = lp * K1;
        #pragma unroll
        for (int i = 0; i < K1; ++i) {
            Ar[2 + 3 * i] = cx[base + i] - pxj;
            Ar[3 + 3 * i] = cy[base + i] - pyj;
            Ar[4 + 3 * i] = nj;
        }
        Ar[53] = 0.0f; Ar[54] = 0.0f; Ar[55] = 0.0f;
    }

    // ---- phase 1b: n_feats gather -> out[..., 0:64] (streamed, non-temporal) ----
    {
        const v4f* f4 = (const v4f*)feats;
        v4f*       o4 = (v4f*)out;
        for (int e = tid; e < ROWS * (Uh / 4); e += 256) {
            int r = e >> 4, c4 = e & 15;
            v4f v = f4[(size_t)(b * Nn + sidx[r]) * (Uh / 4) + c4];
            __builtin_nontemporal_store(
                v, &o4[((size_t)blockStart * K1 + r) * (OUTC / 4) + c4]);
        }
    }

    // ---- phase 1c: ggf (slope + pearson distance), one thread per point ----
    if (tid < PTS) {
        int base = tid * K1;
        float sx = 0.f, sy = 0.f;
        #pragma unroll
        for (int i = 0; i < K1; ++i) { sx += cx[base + i]; sy += cy[base + i]; }
        float mx = sx * (1.0f / K1), my = sy * (1.0f / K1);
        float cov = 0.f, vx = 0.f, vy = 0.f;
        #pragma unroll
        for (int i = 0; i < K1; ++i) {
            float xc = cx[base + i] - mx, yc = cy[base + i] - my;
            cov += xc * yc; vx += xc * xc; vy += yc * yc;
        }
        cov *= (1.0f / K1); vx *= (1.0f / K1); vy *= (1.0f / K1);
        float m    = cov / (vx + EPSF);
        float pear = cov / (sqrtf(vx * vy) + EPSF);
        ggf[(size_t)(blockStart + tid) * 2 + 0] = m;
        ggf[(size_t)(blockStart + tid) * 2 + 1] = 1.0f - pear;
    }
    __syncthreads();

    // ---- phase 2: WMMA  relu(rppe @ W + b) -> out[..., 64:128] ----
    // 17 M-tiles x 4 N-tiles = 68 tile jobs across 8 waves (wave-uniform loop).
    const int w = tid >> 5, lane = tid & 31;
    const int lane15 = lane & 15, hi = lane >> 4;
    for (int job = w; job < MTILE * NTILE; job += 8) {
        int mt = job >> 2, nt = job & 3;
        int Mb = mt * 16, Nb = nt * 16;
        float bv = bias[Nb + lane15];          // C[m][n] = bias[n] (n = lane&15)
        v8f acc = {bv, bv, bv, bv, bv, bv, bv, bv};
        // A: lanes 0-15 row M=Mb+lane, K = 4k..4k+1 ; lanes 16-31 same row, K+2
        const float* Ap = ldsA  + (Mb + lane15) * CHP + hi * 2;
        // B: lanes 0-15 col N=Nb+lane, K = 4k..4k+1 ; lanes 16-31 same col, K+2
        const float* Bp = ldsWt + (Nb + lane15) * CHP + hi * 2;
        #pragma unroll
        for (int kk = 0; kk < KSTEP; ++kk) {
            v2f a  = *(const v2f*)(Ap + kk * 4);
            v2f bb = *(const v2f*)(Bp + kk * 4);
            acc = __builtin_amdgcn_wmma_f32_16x16x4_f32(
                false, a, false, bb, (short)0, acc, false, false);
        }
        // D layout: VGPR v -> M = Mb + v + hi*8, N = Nb + (lane&15)
        size_t rowBase = (size_t)blockStart * K1 + Mb + hi * 8;
        #pragma unroll
        for (int v = 0; v < 8; ++v) {
            float val = acc[v] > 0.0f ? acc[v] : 0.0f;   // fused ReLU
            __builtin_nontemporal_store(
                val, &out[(rowBase + v) * OUTC + Uh + Nb + lane15]);
        }
    }
}

extern "C" void kernel_launch(void* const* d_in, const int* in_sizes, int n_in,
                              void* d_out, int out_size, void* d_ws, size_t ws_size,
                              hipStream_t stream) {
    (void)in_sizes; (void)n_in; (void)out_size; (void)d_ws; (void)ws_size;
    const float* pc    = (const float*)d_in[0];
    const float* feats = (const float*)d_in[1];
    const int*   nidx  = (const int*)d_in[2];
    const float* W     = (const float*)d_in[3];
    const float* bvec  = (const float*)d_in[4];
    float* out = (float*)d_out;
    float* ggf = out + (size_t)Bb * Nn * K1 * OUTC;   // tuple output: out then ggf

    size_t shmem = (size_t)(ROWS * CHP + Uh * CHP + 3 * ROWS) * sizeof(float)
                 + (size_t)ROWS * sizeof(int);        // ~78 KB (WGP has 320 KB LDS)
    hipFuncSetAttribute(reinterpret_cast<const void*>(locse_wmma_kernel),
                        hipFuncAttributeMaxDynamicSharedMemorySize, (int)shmem);

    dim3 grid(Bb * Nn / PTS);   // 4096 blocks of 16 points
    locse_wmma_kernel<<<grid, 256, shmem, stream>>>(pc, feats, nidx, W, bvec, out, ggf);
}